// EntailmentTransformerBlock_23888608100514
// MI455X (gfx1250) — compile-verified
//
#include <hip/hip_runtime.h>

#define E     640
#define L     128
#define HEADS 5
#define HD    128
#define FDIM  2560
#define ROWS_TOTAL 16384   // n*s*L = 64*2*128
#define ROWB  32           // rows per block for the GEMM/LN kernels

typedef float v2f __attribute__((ext_vector_type(2)));
typedef float v8f __attribute__((ext_vector_type(8)));

static __device__ __forceinline__ v8f wmma_f32(v2f a, v2f b, v8f c) {
  // D = A(16x4) * B(4x16) + C(16x16), fp32, wave32
  return __builtin_amdgcn_wmma_f32_16x16x4_f32(false, a, false, b, (short)0, c, false, false);
}

static __device__ __forceinline__ v8f zero8() {
  v8f z;
#pragma unroll
  for (int i = 0; i < 8; ++i) z[i] = 0.0f;
  return z;
}

// --- CDNA5 async global->LDS copies (ASYNCcnt-tracked, no VGPR round-trip) --
// LDS generic address truncated to 32 bits == LDS byte offset (ISA 10.2).
static __device__ __forceinline__ uint32_t lds_off(const void* p) {
  return (uint32_t)(uintptr_t)p;
}
static __device__ __forceinline__ void async_ld_b32(uint32_t lds, uint32_t goff,
                                                    const void* base) {
  asm volatile("global_load_async_to_lds_b32 %0, %1, %2"
               :: "v"(lds), "v"(goff), "s"(base) : "memory");
}
static __device__ __forceinline__ void async_ld_b128(uint32_t lds, uint32_t goff,
                                                     const void* base) {
  asm volatile("global_load_async_to_lds_b128 %0, %1, %2"
               :: "v"(lds), "v"(goff), "s"(base) : "memory");
}
static __device__ __forceinline__ void async_wait0() {
  asm volatile("s_wait_asynccnt 0" ::: "memory");
}

// ---------------------------------------------------------------------------
// Row-block GEMM: OUT[32r x 640] = A[32r x 640] @ W[640 x 640]
// 8 waves; wave w owns 5 col-tiles (col = w*80) x 2 row-tiles.
// One B fragment feeds both row-tiles (halves weight vmem per WMMA).
// ---------------------------------------------------------------------------
__global__ __launch_bounds__(256) void rowgemm_kernel(const float* __restrict__ A,
                                                      const float* __restrict__ W,
                                                      float* __restrict__ OUT) {
  __shared__ float As[ROWB][20];          // 32x16 chunk, pad 20
  const int rowbase = blockIdx.x * ROWB;
  const int tid  = threadIdx.x;
  const int wv   = tid >> 5;
  const int lane = tid & 31;
  const int ln16 = lane & 15;
  const int kh   = lane >> 4;
  const int cw   = wv * 80;
  const int ar   = tid >> 4, ac = tid & 15;   // 16x16 loader coords

  v8f acc[2][5];
#pragma unroll
  for (int rt = 0; rt < 2; ++rt)
#pragma unroll
    for (int t = 0; t < 5; ++t) acc[rt][t] = zero8();

  for (int k0 = 0; k0 < E; k0 += 16) {
    async_ld_b32(lds_off(&As[ar][ac]),
                 (uint32_t)(((rowbase + ar) * E + k0 + ac) * 4), A);
    async_ld_b32(lds_off(&As[ar + 16][ac]),
                 (uint32_t)(((rowbase + ar + 16) * E + k0 + ac) * 4), A);
    async_wait0();
    __syncthreads();
#pragma unroll
    for (int s = 0; s < 4; ++s) {
      const int kl = 4 * s + 2 * kh;
      v2f a0; a0.x = As[ln16][kl];      a0.y = As[ln16][kl + 1];
      v2f a1; a1.x = As[16 + ln16][kl]; a1.y = As[16 + ln16][kl + 1];
      const int kg = k0 + kl;
#pragma unroll
      for (int t = 0; t < 5; ++t) {
        const int col = cw + t * 16 + ln16;
        v2f b; b.x = W[kg * E + col]; b.y = W[(kg + 1) * E + col];
        acc[0][t] = wmma_f32(a0, b, acc[0][t]);
        acc[1][t] = wmma_f32(a1, b, acc[1][t]);
      }
    }
    __syncthreads();
  }
#pragma unroll
  for (int rt = 0; rt < 2; ++rt)
#pragma unroll
    for (int t = 0; t < 5; ++t) {
      const int col = cw + t * 16 + ln16;
#pragma unroll
      for (int r = 0; r < 8; ++r) {
        const int row = rowbase + rt * 16 + r + 8 * kh;
        OUT[row * E + col] = acc[rt][t][r];
      }
    }
}

// ---------------------------------------------------------------------------
// Attention: one block per (n*s, head). L=128, hd=128.
// energy = q @ k^T (scaled 1/sqrt(L)), mask by QUERY row (reference quirk),
// softmax over QUERY axis (= column-wise softmax), out = attn @ v.
// ---------------------------------------------------------------------------
__global__ __launch_bounds__(256) void attn_kernel(const float* __restrict__ Q,
                                                   const float* __restrict__ Km,
                                                   const float* __restrict__ V,
                                                   const int* __restrict__ mask,
                                                   float* __restrict__ O) {
  __shared__ float wstat[8][128];
  __shared__ float attnS[128][68];
  const int bid = blockIdx.x;
  const int h   = bid % HEADS;
  const int ns  = bid / HEADS;
  const float* q = Q + (size_t)ns * (L * E) + h * HD;
  const float* k = Km + (size_t)ns * (L * E) + h * HD;
  const float* v = V + (size_t)ns * (L * E) + h * HD;

  const int tid  = threadIdx.x;
  const int wv   = tid >> 5;
  const int lane = tid & 31;
  const int ln16 = lane & 15;
  const int kh   = lane >> 4;
  const int row0 = wv * 16;

  v8f acc[8];
#pragma unroll
  for (int t = 0; t < 8; ++t) acc[t] = zero8();

  for (int k0 = 0; k0 < HD; k0 += 4) {
    const float* ap = q + (row0 + ln16) * E + k0 + 2 * kh;
    v2f a; a.x = ap[0]; a.y = ap[1];
#pragma unroll
    for (int t = 0; t < 8; ++t) {
      const float* bp = k + (t * 16 + ln16) * E + k0 + 2 * kh;  // B = k^T
      v2f b; b.x = bp[0]; b.y = bp[1];
      acc[t] = wmma_f32(a, b, acc[t]);
    }
  }

  const float rL = 0.0883883476483184f;   // 1/sqrt(128)
#pragma unroll
  for (int r = 0; r < 8; ++r) {
    const int row  = row0 + r + 8 * kh;
    const int mrow = mask[(size_t)ns * (L * E) + row * E];      // mask[..., row, 0]
#pragma unroll
    for (int t = 0; t < 8; ++t) {
      float e = acc[t][r];
      e = (mrow == 0) ? -1e20f : e;
      acc[t][r] = e * rL;
    }
  }

  float cmax[8], csum[8];
#pragma unroll
  for (int t = 0; t < 8; ++t) {
    float lm = acc[t][0];
#pragma unroll
    for (int r = 1; r < 8; ++r) lm = fmaxf(lm, acc[t][r]);
    lm = fmaxf(lm, __shfl_xor(lm, 16, 32));
    if (lane < 16) wstat[wv][t * 16 + ln16] = lm;
  }
  __syncthreads();
#pragma unroll
  for (int t = 0; t < 8; ++t) {
    const int col = t * 16 + ln16;
    float mx = wstat[0][col];
#pragma unroll
    for (int w2 = 1; w2 < 8; ++w2) mx = fmaxf(mx, wstat[w2][col]);
    cmax[t] = mx;
  }
  __syncthreads();
#pragma unroll
  for (int t = 0; t < 8; ++t) {
    float ls = 0.0f;
#pragma unroll
    for (int r = 0; r < 8; ++r) {
      const float e = __expf(acc[t][r] - cmax[t]);
      acc[t][r] = e;
      ls += e;
    }
    ls += __shfl_xor(ls, 16, 32);
    if (lane < 16) wstat[wv][t * 16 + ln16] = ls;
  }
  __syncthreads();
#pragma unroll
  for (int t = 0; t < 8; ++t) {
    const int col = t * 16 + ln16;
    float s = 0.0f;
#pragma unroll
    for (int w2 = 0; w2 < 8; ++w2) s += wstat[w2][col];
    csum[t] = s;
  }
#pragma unroll
  for (int t = 0; t < 8; ++t) {
    const float inv = 1.0f / csum[t];
#pragma unroll
    for (int r = 0; r < 8; ++r) acc[t][r] *= inv;
  }

  v8f oacc[8];
#pragma unroll
  for (int t = 0; t < 8; ++t) oacc[t] = zero8();

  for (int phase = 0; phase < 2; ++phase) {
    __syncthreads();
#pragma unroll
    for (int tt = 0; tt < 4; ++tt) {
      const int t = phase * 4 + tt;
#pragma unroll
      for (int r = 0; r < 8; ++r)
        attnS[row0 + r + 8 * kh][tt * 16 + ln16] = acc[t][r];
    }
    __syncthreads();
    const int kbase = phase * 64;
    for (int k0 = 0; k0 < 64; k0 += 4) {
      v2f a;
      a.x = attnS[row0 + ln16][k0 + 2 * kh];
      a.y = attnS[row0 + ln16][k0 + 2 * kh + 1];
      const int kg = kbase + k0 + 2 * kh;
#pragma unroll
      for (int t = 0; t < 8; ++t) {
        const int col = t * 16 + ln16;
        v2f b; b.x = v[kg * E + col]; b.y = v[(kg + 1) * E + col];
        oacc[t] = wmma_f32(a, b, oacc[t]);
      }
    }
  }

#pragma unroll
  for (int t = 0; t < 8; ++t) {
    const int col = t * 16 + ln16;
#pragma unroll
    for (int r = 0; r < 8; ++r) {
      const int row = row0 + r + 8 * kh;
      O[((size_t)ns * L + row) * E + h * HD + col] = oacc[t][r];
    }
  }
}

// ---------------------------------------------------------------------------
// X = LayerNorm( resid + (Ain @ Wo + bo) ), 32-row blocks, full 640-wide rows.
// ---------------------------------------------------------------------------
__global__ __launch_bounds__(256) void proj_ln_kernel(const float* __restrict__ Ain,
                                                      const float* __restrict__ Wo,
                                                      const float* __restrict__ bo,
                                                      const float* __restrict__ resid,
                                                      const float* __restrict__ g,
                                                      const float* __restrict__ be,
                                                      float* __restrict__ X) {
  __shared__ float As[ROWB][20];
  __shared__ float rsum[ROWB], rsq[ROWB];
  const int rowbase = blockIdx.x * ROWB;
  const int tid  = threadIdx.x;
  const int wv   = tid >> 5;
  const int lane = tid & 31;
  const int ln16 = lane & 15;
  const int kh   = lane >> 4;
  const int cw   = wv * 80;
  const int ar   = tid >> 4, ac = tid & 15;

  v8f acc[2][5];
#pragma unroll
  for (int rt = 0; rt < 2; ++rt)
#pragma unroll
    for (int t = 0; t < 5; ++t) acc[rt][t] = zero8();

  for (int k0 = 0; k0 < E; k0 += 16) {
    async_ld_b32(lds_off(&As[ar][ac]),
                 (uint32_t)(((rowbase + ar) * E + k0 + ac) * 4), Ain);
    async_ld_b32(lds_off(&As[ar + 16][ac]),
                 (uint32_t)(((rowbase + ar + 16) * E + k0 + ac) * 4), Ain);
    async_wait0();
    __syncthreads();
#pragma unroll
    for (int s = 0; s < 4; ++s) {
      const int kl = 4 * s + 2 * kh;
      v2f a0; a0.x = As[ln16][kl];      a0.y = As[ln16][kl + 1];
      v2f a1; a1.x = As[16 + ln16][kl]; a1.y = As[16 + ln16][kl + 1];
      const int kg = k0 + kl;
#pragma unroll
      for (int t = 0; t < 5; ++t) {
        const int col = cw + t * 16 + ln16;
        v2f b; b.x = Wo[kg * E + col]; b.y = Wo[(kg + 1) * E + col];
        acc[0][t] = wmma_f32(a0, b, acc[0][t]);
        acc[1][t] = wmma_f32(a1, b, acc[1][t]);
      }
    }
    __syncthreads();
  }

  // bias + residual
#pragma unroll
  for (int rt = 0; rt < 2; ++rt)
#pragma unroll
    for (int t = 0; t < 5; ++t) {
      const int col = cw + t * 16 + ln16;
      const float bb = bo[col];
#pragma unroll
      for (int r = 0; r < 8; ++r) {
        const int row = rowbase + rt * 16 + r + 8 * kh;
        acc[rt][t][r] = acc[rt][t][r] + bb + resid[row * E + col];
      }
    }

  if (tid < ROWB) { rsum[tid] = 0.0f; rsq[tid] = 0.0f; }
  __syncthreads();
#pragma unroll
  for (int rt = 0; rt < 2; ++rt)
#pragma unroll
    for (int r = 0; r < 8; ++r) {
      float ps = 0.0f, pq = 0.0f;
#pragma unroll
      for (int t = 0; t < 5; ++t) { const float x = acc[rt][t][r]; ps += x; pq += x * x; }
      for (int off = 1; off < 16; off <<= 1) {
        ps += __shfl_xor(ps, off, 32);
        pq += __shfl_xor(pq, off, 32);
      }
      if (ln16 == 0) {
        atomicAdd(&rsum[rt * 16 + r + 8 * kh], ps);
        atomicAdd(&rsq[rt * 16 + r + 8 * kh], pq);
      }
    }
  __syncthreads();

  const float invE = 1.0f / (float)E;
#pragma unroll
  for (int rt = 0; rt < 2; ++rt)
#pragma unroll
    for (int t = 0; t < 5; ++t) {
      const int col = cw + t * 16 + ln16;
      const float gg = g[col], bb = be[col];
#pragma unroll
      for (int r = 0; r < 8; ++r) {
        const int rl  = rt * 16 + r + 8 * kh;
        const int row = rowbase + rl;
        const float mu  = rsum[rl] * invE;
        const float var = rsq[rl] * invE - mu * mu;
        const float inv = rsqrtf(var + 1e-5f);
        X[row * E + col] = (acc[rt][t][r] - mu) * inv * gg + bb;
      }
    }
}

// ---------------------------------------------------------------------------
// OUT = LayerNorm( X + relu(X@W1+b1)@W2 + b2 ), fused, 32-row blocks.
// X block (32x640) LDS-resident via async B128 DMA; F in 128-wide chunks.
// LDS ~100KB (ISA allows up to 320KB/workgroup on gfx1250).
// ---------------------------------------------------------------------------
__global__ __launch_bounds__(256) void ffn_ln_kernel(const float* __restrict__ X,
                                                     const float* __restrict__ W1,
                                                     const float* __restrict__ b1,
                                                     const float* __restrict__ W2,
                                                     const float* __restrict__ b2,
                                                     const float* __restrict__ g,
                                                     const float* __restrict__ be,
                                                     float* __restrict__ OUT) {
  __shared__ float Xs[ROWB][644];         // 32x640, padded (82.4KB)
  __shared__ float Hs[ROWB][132];         // 32x128, padded (16.9KB)
  __shared__ float rsum[ROWB], rsq[ROWB];
  const int rowbase = blockIdx.x * ROWB;
  const int tid  = threadIdx.x;
  const int wv   = tid >> 5;
  const int lane = tid & 31;
  const int ln16 = lane & 15;
  const int kh   = lane >> 4;
  const int cw   = wv * 80;

  // async DMA the X block into LDS, 16B granules
  for (int i = tid; i < ROWB * (E / 4); i += 256) {
    const int r  = i / (E / 4);
    const int c4 = (i % (E / 4)) * 4;
    async_ld_b128(lds_off(&Xs[r][c4]),
                  (uint32_t)(((rowbase + r) * E + c4) * 4), X);
  }
  async_wait0();
  __syncthreads();

  v8f yacc[2][5];
#pragma unroll
  for (int rt = 0; rt < 2; ++rt)
#pragma unroll
    for (int t = 0; t < 5; ++t) yacc[rt][t] = zero8();

  for (int fc = 0; fc < FDIM; fc += 128) {
    // phase 1: two 16x16 h-tiles per wave (rows rt*16.., cols fc + wv*16)
    v8f hacc[2] = {zero8(), zero8()};
    const int colF = fc + wv * 16 + ln16;
    for (int k0 = 0; k0 < E; k0 += 4) {
      const int kg = k0 + 2 * kh;
      v2f a0; a0.x = Xs[ln16][kg];      a0.y = Xs[ln16][kg + 1];
      v2f a1; a1.x = Xs[16 + ln16][kg]; a1.y = Xs[16 + ln16][kg + 1];
      v2f b; b.x = W1[kg * FDIM + colF]; b.y = W1[(kg + 1) * FDIM + colF];
      hacc[0] = wmma_f32(a0, b, hacc[0]);
      hacc[1] = wmma_f32(a1, b, hacc[1]);
    }
    const float bb1 = b1[colF];
#pragma unroll
    for (int rt = 0; rt < 2; ++rt)
#pragma unroll
      for (int r = 0; r < 8; ++r)
        Hs[rt * 16 + r + 8 * kh][wv * 16 + ln16] = fmaxf(hacc[rt][r] + bb1, 0.0f);
    __syncthreads();

    // phase 2: y += h @ W2 (K = 128 chunk); one B fragment feeds 2 row-tiles
    for (int k0 = 0; k0 < 128; k0 += 4) {
      const int kl = k0 + 2 * kh;
      v2f a0; a0.x = Hs[ln16][kl];      a0.y = Hs[ln16][kl + 1];
      v2f a1; a1.x = Hs[16 + ln16][kl]; a1.y = Hs[16 + ln16][kl + 1];
      const int kg = fc + kl;
#pragma unroll
      for (int t = 0; t < 5; ++t) {
        const int col = cw + t * 16 + ln16;
        v2f b; b.x = W2[kg * E + col]; b.y = W2[(kg + 1) * E + col];
        yacc[0][t] = wmma_f32(a0, b, yacc[0][t]);
        yacc[1][t] = wmma_f32(a1, b, yacc[1][t]);
      }
    }
    __syncthreads();
  }

  // bias + residual
#pragma unroll
  for (int rt = 0; rt < 2; ++rt)
#pragma unroll
    for (int t = 0; t < 5; ++t) {
      const int col = cw + t * 16 + ln16;
      const float bb = b2[col];
#pragma unroll
      for (int r = 0; r < 8; ++r) {
        const int rl = rt * 16 + r + 8 * kh;
        yacc[rt][t][r] = yacc[rt][t][r] + bb + Xs[rl][col];
      }
    }

  if (tid < ROWB) { rsum[tid] = 0.0f; rsq[tid] = 0.0f; }
  __syncthreads();
#pragma unroll
  for (int rt = 0; rt < 2; ++rt)
#pragma unroll
    for (int r = 0; r < 8; ++r) {
      float ps = 0.0f, pq = 0.0f;
#pragma unroll
      for (int t = 0; t < 5; ++t) { const float x = yacc[rt][t][r]; ps += x; pq += x * x; }
      for (int off = 1; off < 16; off <<= 1) {
        ps += __shfl_xor(ps, off, 32);
        pq += __shfl_xor(pq, off, 32);
      }
      if (ln16 == 0) {
        atomicAdd(&rsum[rt * 16 + r + 8 * kh], ps);
        atomicAdd(&rsq[rt * 16 + r + 8 * kh], pq);
      }
    }
  __syncthreads();

  const float invE = 1.0f / (float)E;
#pragma unroll
  for (int rt = 0; rt < 2; ++rt)
#pragma unroll
    for (int t = 0; t < 5; ++t) {
      const int col = cw + t * 16 + ln16;
      const float gg = g[col], bb = be[col];
#pragma unroll
      for (int r = 0; r < 8; ++r) {
        const int rl  = rt * 16 + r + 8 * kh;
        const int row = rowbase + rl;
        const float mu  = rsum[rl] * invE;
        const float var = rsq[rl] * invE - mu * mu;
        const float inv = rsqrtf(var + 1e-5f);
        OUT[row * E + col] = (yacc[rt][t][r] - mu) * inv * gg + bb;
      }
    }
}

// ---------------------------------------------------------------------------
extern "C" void kernel_launch(void* const* d_in, const int* in_sizes, int n_in,
                              void* d_out, int out_size, void* d_ws, size_t ws_size,
                              hipStream_t stream) {
  const float* value = (const float*)d_in[0];
  const float* key   = (const float*)d_in[1];
  const float* query = (const float*)d_in[2];
  const int*   mask  = (const int*)d_in[3];
  const float* Wv  = (const float*)d_in[4];
  const float* Wk  = (const float*)d_in[5];
  const float* Wq  = (const float*)d_in[6];
  const float* Wo  = (const float*)d_in[7];
  const float* bo  = (const float*)d_in[8];
  const float* W1  = (const float*)d_in[9];
  const float* b1  = (const float*)d_in[10];
  const float* W2  = (const float*)d_in[11];
  const float* b2  = (const float*)d_in[12];
  const float* g1  = (const float*)d_in[13];
  const float* be1 = (const float*)d_in[14];
  const float* g2  = (const float*)d_in[15];
  const float* be2 = (const float*)d_in[16];
  float* out = (float*)d_out;
  float* ws  = (float*)d_ws;

  const size_t QS = (size_t)ROWS_TOTAL * E;   // 41.9MB per buffer
  float* Qb = ws;
  float* Kb = ws + QS;
  float* Vb = ws + 2 * QS;
  float* Ob = ws + 3 * QS;
  float* Xb = ws + 4 * QS;

  const dim3 blk(256);
  const int rowBlocks = ROWS_TOTAL / ROWB;    // 512

  rowgemm_kernel<<<rowBlocks, blk, 0, stream>>>(query, Wq, Qb);
  rowgemm_kernel<<<rowBlocks, blk, 0, stream>>>(key,   Wk, Kb);
  rowgemm_kernel<<<rowBlocks, blk, 0, stream>>>(value, Wv, Vb);
  attn_kernel<<<128 * HEADS, blk, 0, stream>>>(Qb, Kb, Vb, mask, Ob);
  proj_ln_kernel<<<rowBlocks, blk, 0, stream>>>(Ob, Wo, bo, query, g1, be1, Xb);
  ffn_ln_kernel<<<rowBlocks, blk, 0, stream>>>(Xb, W1, b1, W2, b2, g2, be2, out);
}